// FastCrossEntropyLoss_25237227831298
// MI455X (gfx1250) — compile-verified
//
#include <hip/hip_runtime.h>

#define EPSF 1e-8f

constexpr int B = 512;
constexpr int T = 256;
constexpr int C = 400;
constexpr int TC = 16;                    // time rows per LDS chunk
constexpr int NCHUNK = T / TC;            // 16 chunks
constexpr int NTH = 512;                  // 16 waves of 32
// chunk = TC*C*4 = 25600 B = 1600 B128 transfers; per-thread: 3 (+1 for tid<64)

__device__ __forceinline__ unsigned lds_off_u32(const void* p) {
  // generic pointer to __shared__ -> low 32 bits are the LDS byte offset
  return (unsigned)(unsigned long long)p;
}

// Async DMA of one 25600-B chunk, global -> LDS.
// saddr form: mem = SGPR64 + VGPR32 + INST_OFFSET; lds = VDST + INST_OFFSET
// (per CDNA5 ISA the instruction offset is added to BOTH addresses), so the
// per-lane transfers differ only in the offset immediate: zero per-issue ALU.
__device__ __forceinline__ void async_chunk(const float* __restrict__ gbase,
                                            unsigned lds_base, int tid) {
  const unsigned v = (unsigned)tid * 16u;  // lane byte offset
  const unsigned l = lds_base + v;         // lane LDS dest
  asm volatile("global_load_async_to_lds_b128 %0, %1, %2 offset:0"
               :: "v"(l), "v"(v), "s"(gbase) : "memory");
  asm volatile("global_load_async_to_lds_b128 %0, %1, %2 offset:8192"
               :: "v"(l), "v"(v), "s"(gbase) : "memory");
  asm volatile("global_load_async_to_lds_b128 %0, %1, %2 offset:16384"
               :: "v"(l), "v"(v), "s"(gbase) : "memory");
  if (tid < 64) {  // wave-uniform tail (waves 0-1): 1600 = 3*512 + 64
    asm volatile("global_load_async_to_lds_b128 %0, %1, %2 offset:24576"
                 :: "v"(l), "v"(v), "s"(gbase) : "memory");
  }
}

__device__ __forceinline__ float wave_max(float v) {
#pragma unroll
  for (int off = 16; off > 0; off >>= 1) v = fmaxf(v, __shfl_xor(v, off, 32));
  return v;
}
__device__ __forceinline__ float wave_sum(float v) {
#pragma unroll
  for (int off = 16; off > 0; off >>= 1) v += __shfl_xor(v, off, 32);
  return v;
}

__global__ __launch_bounds__(NTH)
void fastce_main(const float* __restrict__ x, const int* __restrict__ labels,
                 float* __restrict__ out, float* __restrict__ ws) {
  __shared__ __align__(16) float buf[2][TC * C];   // 2 x 25600 B double buffer
  __shared__ float s_inv[TC];                      // per-row 1/sum(exp)

  const int b    = blockIdx.x;
  const int tid  = threadIdx.x;
  const int wave = tid >> 5;
  const int lane = tid & 31;
  const float* xb = x + (size_t)b * T * C;
  const int lab = labels[b];                       // uniform scalar load, hoisted

  // per-class running state (threads 0..C-1 own class c = tid)
  float runcm = 0.f;   // running cummax over time
  float sumcm = 0.f;   // sum of cummax over time
  float maxp  = 0.f;   // max prob over time
  int   arg   = 0;     // argmax (first occurrence)

  // prefetch chunk 0
  async_chunk(xb, lds_off_u32(&buf[0][0]), tid);

  for (int k = 0; k < NCHUNK; ++k) {
    const int bk = k & 1;
    if (k + 1 < NCHUNK) {
      // overlap next chunk's DMA with this chunk's compute
      async_chunk(xb + (size_t)(k + 1) * TC * C, lds_off_u32(&buf[bk ^ 1][0]), tid);
      // async loads retire in order and each wave issued >=3 for chunk k+1,
      // so asynccnt<=3 guarantees all of chunk k has landed in LDS.
      asm volatile("s_wait_asynccnt 0x3" ::: "memory");
    } else {
      asm volatile("s_wait_asynccnt 0x0" ::: "memory");
    }
    __syncthreads();

    // phase A: per time-row softmax stats; rewrite row as exp(x - m) in place
    {
      float* row = &buf[bk][wave * C];
      float mx = -3.402823466e38f;
      for (int c = lane; c < C; c += 32) mx = fmaxf(mx, row[c]);
      mx = wave_max(mx);
      float s = 0.f;
      for (int c = lane; c < C; c += 32) {
        float e = __expf(row[c] - mx);
        row[c] = e;
        s += e;
      }
      s = wave_sum(s);
      if (lane == 0) s_inv[wave] = 1.0f / s;
    }
    __syncthreads();

    // phase B: per-class sequential prefix scan over the 16 time rows
    if (tid < C) {
#pragma unroll
      for (int t = 0; t < TC; ++t) {
        float p = buf[bk][t * C + tid] * s_inv[t];
        p = fmaxf(p, EPSF);                 // jnp.maximum(softmax, EPS)
        runcm = fmaxf(runcm, p);
        sumcm += runcm;
        const int tt = k * TC + t;
        if (p > maxp) { maxp = p; arg = tt; }   // strict > => first occurrence
      }
    }
    __syncthreads();   // all reads of buf[bk] done before next DMA overwrites it
  }

  if (tid < C) {
    const size_t o = (size_t)b * C + tid;
    float* out_idx = out + 1;
    float* out_mp  = out + 1 + (size_t)B * C;
    float* out_ex  = out + 1 + 2 * (size_t)B * C;
    out_idx[o] = (float)arg;
    out_mp[o]  = maxp;                       // already >= EPS
    const float ex = (float)T - sumcm / maxp;
    out_ex[o] = ex;
    if (tid == lab) {
      // LAMBDA == 0: log(LAMBDA + (1-LAMBDA)*i_x + EPS) = log(e_x/T + EPS)
      ws[b] = -__logf(maxp) + __logf(ex * (1.0f / (float)T) + EPSF);
    }
  }
}

__global__ __launch_bounds__(512)
void fastce_reduce(const float* __restrict__ ws, float* __restrict__ out) {
  __shared__ float s[512];
  const int t = threadIdx.x;
  s[t] = ws[t];
  __syncthreads();
  for (int off = 256; off > 0; off >>= 1) {
    if (t < off) s[t] += s[t + off];
    __syncthreads();
  }
  if (t == 0) out[0] = s[0] * (1.0f / (float)B);
}

extern "C" void kernel_launch(void* const* d_in, const int* in_sizes, int n_in,
                              void* d_out, int out_size, void* d_ws, size_t ws_size,
                              hipStream_t stream) {
  const float* x      = (const float*)d_in[0];
  const int*   labels = (const int*)d_in[1];
  float* out = (float*)d_out;
  float* ws  = (float*)d_ws;   // needs B*4 = 2 KB

  hipLaunchKernelGGL(fastce_main, dim3(B), dim3(NTH), 0, stream, x, labels, out, ws);
  hipLaunchKernelGGL(fastce_reduce, dim3(1), dim3(512), 0, stream, ws, out);
}